// GNNRouteEncoder_85298050499128
// MI455X (gfx1250) — compile-verified
//
#include <hip/hip_runtime.h>
#include <math.h>

// ---------------------------------------------------------------------------
// GNN route encoder forward for MI455X (gfx1250, wave32).
// All dense matmuls use V_WMMA_F32_16X16X4_F32 (exact fp32 semantics matching
// the fp32 JAX reference) with LDS-staged, double-buffered tiles fed by the
// CDNA5 async global->LDS copy path when the toolchain exposes it.
// ---------------------------------------------------------------------------

typedef __attribute__((ext_vector_type(2))) float v2f;
typedef __attribute__((ext_vector_type(8))) float v8f;

#define HID     256
#define HEADS   8
#define CDIM    32      // HID / HEADS
#define NLAYERS 2
#define NB      32      // batch
#define SMAX    512
#define RMAX    256
#define NEDGE   1024
#define PROOUT  32
#define FCIN    (HID + PROOUT)

#if defined(__has_builtin)
#if __has_builtin(__builtin_amdgcn_global_load_async_to_lds_b128) && \
    __has_builtin(__builtin_amdgcn_s_wait_asynccnt)
#define USE_ASYNC_COPY 1
#endif
#endif
#ifndef USE_ASYNC_COPY
#define USE_ASYNC_COPY 0
#endif

#if USE_ASYNC_COPY
typedef int v4i_vec __attribute__((vector_size(16)));
typedef __attribute__((address_space(1))) v4i_vec* g_v4i_ptr;   // global (AS1)
typedef __attribute__((address_space(3))) v4i_vec* s_v4i_ptr;   // LDS (AS3)
#endif

// 16-byte global -> LDS copy (async DMA if available, else through VGPRs)
__device__ __forceinline__ void cp16_g2s(float* s, const float* g) {
#if USE_ASYNC_COPY
    __builtin_amdgcn_global_load_async_to_lds_b128(
        (g_v4i_ptr)(void*)g, (s_v4i_ptr)(void*)s, 0, 0);
#else
    *(float4*)s = *(const float4*)g;
#endif
}
__device__ __forceinline__ void stage_wait() {
#if USE_ASYNC_COPY
    __builtin_amdgcn_s_wait_asynccnt(0);
#endif
    __syncthreads();
}

// -------------------------- wave32 reductions ------------------------------
__device__ __forceinline__ float wred_sum(float v) {
#pragma unroll
    for (int off = 16; off > 0; off >>= 1) v += __shfl_xor(v, off, 32);
    return v;
}
__device__ __forceinline__ float wred_max(float v) {
#pragma unroll
    for (int off = 16; off > 0; off >>= 1) v = fmaxf(v, __shfl_xor(v, off, 32));
    return v;
}
__device__ __forceinline__ void atomicMaxF(float* addr, float val) {
    unsigned int* ua = (unsigned int*)addr;
    unsigned int old = *ua;
    while (true) {
        float cur = __uint_as_float(old);
        if (cur >= val) break;
        unsigned int prev = atomicCAS(ua, old, __float_as_uint(val));
        if (prev == old) break;
        old = prev;
    }
}

// ---------------------------------------------------------------------------
// GEMM: C[M,N] = act(A[M,K] @ W[K,N] + bias), row-major, fp32 WMMA.
// Block: 256 threads = 8 waves (4 M-tiles x 2 N-tile-pairs), tile 64x64,
// K staged through double-buffered LDS in chunks of 32.
// LDS strides padded: A=36 (lane halves hit banks {0,1 mod 4} vs {2,3 mod 4}),
// B=72 (2*72 mod 64 == 16: lane halves use disjoint bank ranges); both keep
// rows 16B-aligned for b128 copies.
// Requires: M % 64 == 0; N, K compile-time in {256,512}.
// ---------------------------------------------------------------------------
template<int N, int K, int ACT>
__global__ __launch_bounds__(256)
void gemm_lds_kernel(const float* __restrict__ A, const float* __restrict__ W,
                     const float* __restrict__ bias, float* __restrict__ C, int M) {
    constexpr int BM = 64, BN = 64, BK = 32;
    constexpr int LDA = 36;
    constexpr int LDB = 72;
    __shared__ float sA[2][BM * LDA];
    __shared__ float sB[2][BK * LDB];

    const int tid  = threadIdx.x;
    const int lane = tid & 31;
    const int wave = tid >> 5;
    const int half = lane >> 4;
    const int l15  = lane & 15;
    const int wm   = wave >> 1;          // 0..3 -> M sub-tile
    const int wn   = wave & 1;           // 0..1 -> N sub-tile pair
    const int gm0  = blockIdx.y * BM;
    const int gn0  = blockIdx.x * BN;

    // copy assignment: 2 x 16B segments per thread for each of A and B
    const int s0 = tid * 2, s1 = tid * 2 + 1;
    const int a_r0 = s0 >> 3, a_k0 = (s0 & 7) * 4;   // A tile 64x32
    const int a_r1 = s1 >> 3, a_k1 = (s1 & 7) * 4;
    const int b_r0 = s0 >> 4, b_c0 = (s0 & 15) * 4;  // B tile 32x64
    const int b_r1 = s1 >> 4, b_c1 = (s1 & 15) * 4;

    const float* gA = A + (size_t)gm0 * K;
    const float* gB = W + gn0;

    auto stage = [&](int buf, int kc) {
        cp16_g2s(&sA[buf][a_r0 * LDA + a_k0], gA + (size_t)a_r0 * K + kc + a_k0);
        cp16_g2s(&sA[buf][a_r1 * LDA + a_k1], gA + (size_t)a_r1 * K + kc + a_k1);
        cp16_g2s(&sB[buf][b_r0 * LDB + b_c0], gB + (size_t)(kc + b_r0) * N + b_c0);
        cp16_g2s(&sB[buf][b_r1 * LDB + b_c1], gB + (size_t)(kc + b_r1) * N + b_c1);
    };

    stage(0, 0);
    stage_wait();

    v8f acc0 = {}, acc1 = {};
    const int m0 = wm * 16;
    const int n0 = wn * 32;

    for (int kc = 0; kc < K; kc += BK) {
        const int buf = (kc / BK) & 1;
        if (kc + BK < K) stage(buf ^ 1, kc + BK);   // overlap next chunk DMA
        const float* As = sA[buf];
        const float* Bs = sB[buf];
#pragma unroll
        for (int kk = 0; kk < BK; kk += 4) {
            const float* ap = As + (m0 + l15) * LDA + kk + (half << 1);
            v2f av;
            av.x = ap[0];
            av.y = ap[1];
            const float* bp = Bs + (kk + (half << 1)) * LDB + n0 + l15;
            v2f b0, b1;
            b0.x = bp[0];        b0.y = bp[LDB];
            b1.x = bp[16];       b1.y = bp[LDB + 16];
            acc0 = __builtin_amdgcn_wmma_f32_16x16x4_f32(false, av, false, b0,
                                                         (short)0, acc0, false, false);
            acc1 = __builtin_amdgcn_wmma_f32_16x16x4_f32(false, av, false, b1,
                                                         (short)0, acc1, false, false);
        }
        stage_wait();
    }

    const int gcol = gn0 + n0 + l15;
    const float bv0 = bias ? bias[gcol] : 0.0f;
    const float bv1 = bias ? bias[gcol + 16] : 0.0f;
    float* cp = C + (size_t)(gm0 + m0 + (half << 3)) * N + gcol;  // D: VGPR i -> row 8*half+i
#pragma unroll
    for (int i = 0; i < 8; ++i) {
        float v0 = acc0[i] + bv0;
        float v1 = acc1[i] + bv1;
        if (ACT) { v0 = fmaxf(v0, 0.0f); v1 = fmaxf(v1, 0.0f); }
        cp[(size_t)i * N] = v0;
        cp[(size_t)i * N + 16] = v1;
    }
}

// ---------------------------------------------------------------------------
// Fused masked attention for one (batch, head, 16-query tile).
// Scores (WMMA, K=32) -> LDS -> stable softmax -> P@V (WMMA, K split 8 ways).
// Lk must be 512 (true for both uses), Lq % 16 == 0.
// ---------------------------------------------------------------------------
__global__ __launch_bounds__(256)
void attention_kernel(const float* __restrict__ Q, const float* __restrict__ Km,
                      const float* __restrict__ Vm, float* __restrict__ O,
                      const int* __restrict__ qlen, const int* __restrict__ klen,
                      int Lq, int Lk) {
    __shared__ float sS[16][SMAX];          // 32 KB score strip
    __shared__ float sP[8][16][CDIM];       // 16 KB PV partials
    const int lane = threadIdx.x & 31;
    const int wave = threadIdx.x >> 5;
    const int half = lane >> 4;
    const int l15  = lane & 15;
    const int q0 = blockIdx.x * 16;
    const int h  = blockIdx.y;
    const int b  = blockIdx.z;
    const int ql = qlen[b], kl = klen[b];
    const float scale = 0.17677669529663687f;   // 1/sqrt(32)

    const float* Qb = Q + (size_t)b * Lq * HID + h * CDIM;
    const float* Kb = Km + (size_t)b * Lk * HID + h * CDIM;
    const float* Vb = Vm + (size_t)b * Lk * HID + h * CDIM;

    // ---- scores: wave w handles key tiles w, w+8, w+16, w+24 ----
    const float* qp = Qb + (size_t)(q0 + l15) * HID + (half << 1);
    for (int kt = wave; kt < Lk / 16; kt += 8) {
        const int n0 = kt * 16;
        const float* kp = Kb + (size_t)(n0 + l15) * HID + (half << 1);
        v8f acc = {};
#pragma unroll
        for (int c0 = 0; c0 < CDIM; c0 += 4) {
            v2f av, bv;
            av.x = qp[c0];  av.y = qp[c0 + 1];
            bv.x = kp[c0];  bv.y = kp[c0 + 1];
            acc = __builtin_amdgcn_wmma_f32_16x16x4_f32(false, av, false, bv,
                                                        (short)0, acc, false, false);
        }
        const int row0 = half * 8;
        const int key  = n0 + l15;
#pragma unroll
        for (int i = 0; i < 8; ++i) {
            const int qrow = q0 + row0 + i;
            sS[row0 + i][key] = (qrow < ql && key < kl) ? acc[i] * scale : -1e9f;
        }
    }
    __syncthreads();

    // ---- stable softmax: wave w owns rows 2w, 2w+1 ----
    for (int r = wave * 2; r < wave * 2 + 2; ++r) {
        float mx = -1e30f;
        for (int k = lane; k < Lk; k += 32) mx = fmaxf(mx, sS[r][k]);
        mx = wred_max(mx);
        float sm = 0.f;
        for (int k = lane; k < Lk; k += 32) {
            float e = expf(sS[r][k] - mx);
            sS[r][k] = e;
            sm += e;
        }
        sm = wred_sum(sm);
        const float inv = 1.0f / sm;
        for (int k = lane; k < Lk; k += 32) sS[r][k] *= inv;
    }
    __syncthreads();

    // ---- P @ V: wave w covers keys [w*Lk/8, (w+1)*Lk/8), both c-tiles ----
    v8f acc0 = {}, acc1 = {};
    const int kbase = wave * (Lk / 8);
    for (int k0 = kbase; k0 < kbase + Lk / 8; k0 += 4) {
        v2f av;
        av.x = sS[l15][k0 + (half << 1)];
        av.y = sS[l15][k0 + (half << 1) + 1];
        const float* vp = Vb + (size_t)(k0 + (half << 1)) * HID;
        v2f b0, b1;
        b0.x = vp[l15];       b0.y = vp[HID + l15];
        b1.x = vp[16 + l15];  b1.y = vp[HID + 16 + l15];
        acc0 = __builtin_amdgcn_wmma_f32_16x16x4_f32(false, av, false, b0,
                                                     (short)0, acc0, false, false);
        acc1 = __builtin_amdgcn_wmma_f32_16x16x4_f32(false, av, false, b1,
                                                     (short)0, acc1, false, false);
    }
    {
        const int row0 = half * 8;
#pragma unroll
        for (int i = 0; i < 8; ++i) {
            sP[wave][row0 + i][l15]      = acc0[i];
            sP[wave][row0 + i][16 + l15] = acc1[i];
        }
    }
    __syncthreads();
    for (int idx = threadIdx.x; idx < 16 * CDIM; idx += 256) {
        const int r = idx >> 5, c = idx & 31;
        float s = 0.f;
#pragma unroll
        for (int w = 0; w < 8; ++w) s += sP[w][r][c];
        O[((size_t)b * Lq + q0 + r) * HID + h * CDIM + c] = s;
    }
}

// ---------------------------------------------------------------------------
// out = a * (x+y - mu) / (std_unbiased + 1e-6) + b   (reference _norm)
// ---------------------------------------------------------------------------
__global__ __launch_bounds__(256)
void add_norm_kernel(const float* __restrict__ X, const float* __restrict__ Y,
                     const float* __restrict__ ga, const float* __restrict__ gb,
                     float* __restrict__ O, int rows) {
    const int lane = threadIdx.x & 31;
    const int wave = threadIdx.x >> 5;
    const int row = blockIdx.x * 8 + wave;
    if (row >= rows) return;
    const float* xp = X + (size_t)row * HID;
    const float* yp = Y + (size_t)row * HID;
    float v[8];
    float s = 0.f;
#pragma unroll
    for (int i = 0; i < 8; ++i) { v[i] = xp[lane + 32 * i] + yp[lane + 32 * i]; s += v[i]; }
    const float mu = wred_sum(s) * (1.0f / HID);
    float ss = 0.f;
#pragma unroll
    for (int i = 0; i < 8; ++i) { float d = v[i] - mu; ss += d * d; }
    ss = wred_sum(ss);
    const float inv = 1.0f / (sqrtf(ss * (1.0f / (HID - 1))) + 1e-6f);
    float* op = O + (size_t)row * HID;
#pragma unroll
    for (int i = 0; i < 8; ++i) {
        const int d = lane + 32 * i;
        op[d] = ga[d] * (v[i] - mu) * inv + gb[d];
    }
}

// ------------------------------- GAT pieces --------------------------------
__global__ void gat_alar_kernel(const float* __restrict__ Xg,
                                const float* __restrict__ a_src,
                                const float* __restrict__ a_dst,
                                float* __restrict__ al, float* __restrict__ ar, int Nn) {
    const int i = blockIdx.x * blockDim.x + threadIdx.x;
    if (i >= Nn * HEADS) return;
    const int h = i & 7, n = i >> 3;
    const float* xp = Xg + (size_t)n * HID + h * CDIM;
    float sl = 0.f, sr = 0.f;
#pragma unroll
    for (int c = 0; c < CDIM; ++c) {
        const float x = xp[c];
        sl += x * a_src[h * CDIM + c];
        sr += x * a_dst[h * CDIM + c];
    }
    al[i] = sl;
    ar[i] = sr;
}

__global__ void gat_init_kernel(float* __restrict__ agg, float* __restrict__ denom,
                                float* __restrict__ mx, int Nn) {
    const int i = blockIdx.x * blockDim.x + threadIdx.x;
    if (i < Nn * HID) agg[i] = 0.f;
    if (i < Nn * HEADS) { denom[i] = 0.f; mx[i] = -1e30f; }
}

__global__ void gat_edge1_kernel(const int* __restrict__ adj,
                                 const float* __restrict__ al, const float* __restrict__ ar,
                                 float* __restrict__ ev, float* __restrict__ mx) {
    const int i = blockIdx.x * blockDim.x + threadIdx.x;   // (edge, head)
    if (i >= NB * NEDGE * HEADS) return;
    const int h = i & 7, e = i >> 3;
    const int b = e / NEDGE, j = e % NEDGE;
    const int src = adj[(size_t)b * 2 * NEDGE + j] + b * RMAX;
    const int dst = adj[(size_t)b * 2 * NEDGE + NEDGE + j] + b * RMAX;
    float v = al[src * HEADS + h] + ar[dst * HEADS + h];
    v = (v >= 0.f) ? v : 0.2f * v;          // leaky_relu(., 0.2)
    ev[i] = v;
    atomicMaxF(&mx[dst * HEADS + h], v);
}

__global__ void gat_edge2_kernel(const int* __restrict__ adj,
                                 const float* __restrict__ ev, const float* __restrict__ mx,
                                 const float* __restrict__ Xg,
                                 float* __restrict__ denom, float* __restrict__ agg) {
    const int i = blockIdx.x * blockDim.x + threadIdx.x;   // (edge, head)
    if (i >= NB * NEDGE * HEADS) return;
    const int h = i & 7, e = i >> 3;
    const int b = e / NEDGE, j = e % NEDGE;
    const int src = adj[(size_t)b * 2 * NEDGE + j] + b * RMAX;
    const int dst = adj[(size_t)b * 2 * NEDGE + NEDGE + j] + b * RMAX;
    const float ex = expf(ev[i] - mx[dst * HEADS + h]);
    atomicAdd(&denom[dst * HEADS + h], ex);
    const float* xp = Xg + (size_t)src * HID + h * CDIM;
    float* ap = agg + (size_t)dst * HID + h * CDIM;
#pragma unroll
    for (int c = 0; c < CDIM; ++c) atomicAdd(&ap[c], ex * xp[c]);
}

// h_new = agg/max(denom,1e-16)+gat_b; ln = LayerNorm(h+h_new, lw, lb);
// gnn = valid ? ln : 0;  route_out = _norm(h + gnn, n1a, n1b)
__global__ __launch_bounds__(256)
void gat_finalize_kernel(const float* __restrict__ h_in, const float* __restrict__ agg,
                         const float* __restrict__ denom, const float* __restrict__ gat_b,
                         const float* __restrict__ lw, const float* __restrict__ lb,
                         const float* __restrict__ n1a, const float* __restrict__ n1b,
                         const int* __restrict__ route_len,
                         float* __restrict__ route_out, int Nn) {
    const int lane = threadIdx.x & 31;
    const int wave = threadIdx.x >> 5;
    const int n = blockIdx.x * 8 + wave;
    if (n >= Nn) return;
    const int b = n / RMAX, r = n % RMAX;
    const float* hp = h_in + (size_t)n * HID;
    const float* ap = agg + (size_t)n * HID;
    float hr[8], t[8];
    float s = 0.f;
#pragma unroll
    for (int i = 0; i < 8; ++i) {
        const int d = lane + 32 * i;
        const int h = d >> 5;
        const float dn = fmaxf(denom[n * HEADS + h], 1e-16f);
        hr[i] = hp[d];
        t[i] = hr[i] + ap[d] / dn + gat_b[d];
        s += t[i];
    }
    const float mu = wred_sum(s) * (1.0f / HID);
    float ss = 0.f;
#pragma unroll
    for (int i = 0; i < 8; ++i) { const float d = t[i] - mu; ss += d * d; }
    ss = wred_sum(ss);
    const float rinv = rsqrtf(ss * (1.0f / HID) + 1e-5f);     // biased LayerNorm
    const bool valid = r < route_len[b];
    float u[8];
    float s2 = 0.f;
#pragma unroll
    for (int i = 0; i < 8; ++i) {
        const int d = lane + 32 * i;
        const float ln = (t[i] - mu) * rinv * lw[d] + lb[d];
        u[i] = hr[i] + (valid ? ln : 0.f);
        s2 += u[i];
    }
    const float mu2 = wred_sum(s2) * (1.0f / HID);
    float ss2 = 0.f;
#pragma unroll
    for (int i = 0; i < 8; ++i) { const float d = u[i] - mu2; ss2 += d * d; }
    ss2 = wred_sum(ss2);
    const float inv2 = 1.0f / (sqrtf(ss2 * (1.0f / (HID - 1))) + 1e-6f);  // unbiased _norm
    float* op = route_out + (size_t)n * HID;
#pragma unroll
    for (int i = 0; i < 8; ++i) {
        const int d = lane + 32 * i;
        op[d] = n1a[d] * (u[i] - mu2) * inv2 + n1b[d];
    }
}

// ------------------------------ data movers --------------------------------
__global__ void transpose_lbd_kernel(const float* __restrict__ in,
                                     float* __restrict__ out, int L) {
    const size_t i = (size_t)blockIdx.x * blockDim.x + threadIdx.x;   // [L,NB,HID] -> [NB,L,HID]
    if (i >= (size_t)L * NB * HID) return;
    const int d = (int)(i % HID);
    const size_t t = i / HID;
    const int b = (int)(t % NB);
    const int l = (int)(t / NB);
    out[((size_t)b * L + l) * HID + d] = in[i];
}

__global__ void write_route_outputs_kernel(const float* __restrict__ re,
                                           float* __restrict__ out) {
    const size_t i = (size_t)blockIdx.x * blockDim.x + threadIdx.x;   // [NB,RMAX,HID] -> [RMAX,NB,HID]
    if (i >= (size_t)NB * RMAX * HID) return;
    const int d = (int)(i % HID);
    const size_t t = i / HID;
    const int r = (int)(t % RMAX);
    const int b = (int)(t / RMAX);
    out[((size_t)r * NB + b) * HID + d] = re[i];
}

// masked mean over route, concat temporal feature, fc + tanh
__global__ __launch_bounds__(HID)
void pool_fc_kernel(const float* __restrict__ re, const int* __restrict__ route_len,
                    const int* __restrict__ pro, const float* __restrict__ temporal,
                    const float* __restrict__ fcW, const float* __restrict__ fcb,
                    float* __restrict__ out) {
    __shared__ float cat[FCIN];
    const int b = blockIdx.x;
    const int t = threadIdx.x;
    if (t < PROOUT) cat[t] = temporal[(size_t)pro[b] * PROOUT + t];
    const int len = route_len[b];
    const float* rp = re + (size_t)b * RMAX * HID + t;
    float s = 0.f;
    for (int r = 0; r < len; ++r) s += rp[(size_t)r * HID];
    cat[PROOUT + t] = s / (float)len;
    __syncthreads();
    float acc = fcb[t];
    for (int i = 0; i < FCIN; ++i) acc += cat[i] * fcW[(size_t)i * HID + t];
    out[(size_t)b * HID + t] = tanhf(acc);
}

// ---------------------------------------------------------------------------
// Host driver. Input order = setup_inputs() dict insertion order, with the
// nested `params` dict flattened in ITS insertion order (indices 6..46).
// ---------------------------------------------------------------------------
extern "C" void kernel_launch(void* const* d_in, const int* in_sizes, int n_in,
                              void* d_out, int out_size, void* d_ws, size_t ws_size,
                              hipStream_t stream) {
    (void)in_sizes; (void)n_in; (void)out_size; (void)ws_size;

    const float* src       = (const float*)d_in[0];   // [SMAX,NB,HID]
    const float* route     = (const float*)d_in[1];   // [RMAX,NB,HID]
    const int*   src_len   = (const int*)d_in[2];
    const int*   route_len = (const int*)d_in[3];
    const int*   adj       = (const int*)d_in[4];     // [NB,2,NEDGE]
    const int*   pro       = (const int*)d_in[5];
    // params (insertion order)
    const float* gps_Wq = (const float*)d_in[6];
    const float* gps_Wk = (const float*)d_in[7];
    const float* gps_Wv = (const float*)d_in[8];
    const float* gps_Wo = (const float*)d_in[9];
    const float* gps_bq = (const float*)d_in[10];
    const float* gps_bk = (const float*)d_in[11];
    const float* gps_bv = (const float*)d_in[12];
    const float* gps_bo = (const float*)d_in[13];
    const float* x_Wq   = (const float*)d_in[14];
    const float* x_Wk   = (const float*)d_in[15];
    const float* x_Wv   = (const float*)d_in[16];
    const float* x_Wo   = (const float*)d_in[17];
    const float* x_bq   = (const float*)d_in[18];
    const float* x_bk   = (const float*)d_in[19];
    const float* x_bv   = (const float*)d_in[20];
    const float* x_bo   = (const float*)d_in[21];
    const float* gps_n1a = (const float*)d_in[22];
    const float* gps_n2a = (const float*)d_in[23];
    const float* rn1a    = (const float*)d_in[24];
    const float* rn2a    = (const float*)d_in[25];
    const float* gat_lw  = (const float*)d_in[26];
    const float* gps_n1b = (const float*)d_in[27];
    const float* gps_n2b = (const float*)d_in[28];
    const float* rn1b    = (const float*)d_in[29];
    const float* rn2b    = (const float*)d_in[30];
    const float* gat_lb  = (const float*)d_in[31];
    const float* gps_fW1 = (const float*)d_in[32];
    const float* gps_fb1 = (const float*)d_in[33];
    const float* gps_fW2 = (const float*)d_in[34];
    const float* gps_fb2 = (const float*)d_in[35];
    const float* rfW1    = (const float*)d_in[36];
    const float* rfb1    = (const float*)d_in[37];
    const float* rfW2    = (const float*)d_in[38];
    const float* rfb2    = (const float*)d_in[39];
    const float* gatW    = (const float*)d_in[40];
    const float* gat_as  = (const float*)d_in[41];
    const float* gat_ad  = (const float*)d_in[42];
    const float* gat_b   = (const float*)d_in[43];
    const float* temporal = (const float*)d_in[44];
    const float* fcW      = (const float*)d_in[45];
    const float* fcb      = (const float*)d_in[46];

    // --------- workspace layout (floats), ~145 MB total ---------
    float* ws = (float*)d_ws;
    size_t o = 0;
    const size_t SG = (size_t)NB * SMAX * HID;   // 4,194,304
    const size_t SR = (size_t)NB * RMAX * HID;   // 2,097,152
    const int    Nn = NB * RMAX;                 // 8192 nodes
    float* gps_emb   = ws + o; o += SG;
    float* route_emb = ws + o; o += SR;
    float* tQ = ws + o; o += SG;
    float* tK = ws + o; o += SG;
    float* tV = ws + o; o += SG;
    float* tO = ws + o; o += SG;
    float* tM = ws + o; o += 2 * SG;             // FFN mid / attention context
    float* route_out  = ws + o; o += SR;
    float* route_out2 = ws + o; o += SR;
    float* al  = ws + o; o += (size_t)Nn * HEADS;
    float* ar  = ws + o; o += (size_t)Nn * HEADS;
    float* ev  = ws + o; o += (size_t)NB * NEDGE * HEADS;
    float* gmx = ws + o; o += (size_t)Nn * HEADS;
    float* gdn = ws + o; o += (size_t)Nn * HEADS;

    const int MG = NB * SMAX;    // 16384 gps rows
    const int MR = NB * RMAX;    // 8192 route rows

    auto gemm = [&](const float* A, const float* W, const float* bias, float* C,
                    int M, int N, int K, int act) {
        dim3 g(N / 64, M / 64);
        if (N == 256 && K == 256) {
            if (act) gemm_lds_kernel<256, 256, 1><<<g, 256, 0, stream>>>(A, W, bias, C, M);
            else     gemm_lds_kernel<256, 256, 0><<<g, 256, 0, stream>>>(A, W, bias, C, M);
        } else if (N == 512 && K == 256) {
            if (act) gemm_lds_kernel<512, 256, 1><<<g, 256, 0, stream>>>(A, W, bias, C, M);
            else     gemm_lds_kernel<512, 256, 0><<<g, 256, 0, stream>>>(A, W, bias, C, M);
        } else {   // N == 256, K == 512
            if (act) gemm_lds_kernel<256, 512, 1><<<g, 256, 0, stream>>>(A, W, bias, C, M);
            else     gemm_lds_kernel<256, 512, 0><<<g, 256, 0, stream>>>(A, W, bias, C, M);
        }
    };

    // seq-major -> batch-major
    transpose_lbd_kernel<<<(int)((SG + 255) / 256), 256, 0, stream>>>(src, gps_emb, SMAX);
    transpose_lbd_kernel<<<(int)((SR + 255) / 256), 256, 0, stream>>>(route, route_emb, RMAX);

    for (int l = 0; l < NLAYERS; ++l) {
        const size_t oW = (size_t)l * HID * HID;
        const size_t oB = (size_t)l * HID;
        const size_t oF = (size_t)l * HID * 2 * HID;
        const size_t oB2 = (size_t)l * 2 * HID;

        // ---- GPS self-attention ----
        gemm(gps_emb, gps_Wq + oW, gps_bq + oB, tQ, MG, HID, HID, 0);
        gemm(gps_emb, gps_Wk + oW, gps_bk + oB, tK, MG, HID, HID, 0);
        gemm(gps_emb, gps_Wv + oW, gps_bv + oB, tV, MG, HID, HID, 0);
        {
            dim3 ga(SMAX / 16, HEADS, NB);
            attention_kernel<<<ga, 256, 0, stream>>>(tQ, tK, tV, tM,
                                                     src_len, src_len, SMAX, SMAX);
        }
        gemm(tM, gps_Wo + oW, gps_bo + oB, tO, MG, HID, HID, 0);
        add_norm_kernel<<<MG / 8, 256, 0, stream>>>(gps_emb, tO, gps_n1a + oB,
                                                    gps_n1b + oB, gps_emb, MG);
        // ---- GPS FFN ----
        gemm(gps_emb, gps_fW1 + oF, gps_fb1 + oB2, tM, MG, 2 * HID, HID, 1);
        gemm(tM, gps_fW2 + oF, gps_fb2 + oB, tO, MG, HID, 2 * HID, 0);
        add_norm_kernel<<<MG / 8, 256, 0, stream>>>(gps_emb, tO, gps_n2a + oB,
                                                    gps_n2b + oB, gps_emb, MG);

        // ---- GAT on route graph ----
        gemm(route_emb, gatW + oW, nullptr, tQ, Nn, HID, HID, 0);   // Xg = h @ W
        gat_alar_kernel<<<(Nn * HEADS + 255) / 256, 256, 0, stream>>>(
            tQ, gat_as + (size_t)l * HEADS * CDIM, gat_ad + (size_t)l * HEADS * CDIM,
            al, ar, Nn);
        gat_init_kernel<<<(Nn * HID + 255) / 256, 256, 0, stream>>>(tK, gdn, gmx, Nn);
        gat_edge1_kernel<<<(NB * NEDGE * HEADS + 255) / 256, 256, 0, stream>>>(
            adj, al, ar, ev, gmx);
        gat_edge2_kernel<<<(NB * NEDGE * HEADS + 255) / 256, 256, 0, stream>>>(
            adj, ev, gmx, tQ, gdn, tK);
        gat_finalize_kernel<<<Nn / 8, 256, 0, stream>>>(
            route_emb, tK, gdn, gat_b + oB, gat_lw + oB, gat_lb + oB,
            rn1a + oB, rn1b + oB, route_len, route_out, Nn);

        // ---- cross attention route -> gps ----
        gemm(route_out, x_Wq + oW, x_bq + oB, tQ, MR, HID, HID, 0);
        gemm(gps_emb, x_Wk + oW, x_bk + oB, tK, MG, HID, HID, 0);
        gemm(gps_emb, x_Wv + oW, x_bv + oB, tV, MG, HID, HID, 0);
        {
            dim3 ga(RMAX / 16, HEADS, NB);
            attention_kernel<<<ga, 256, 0, stream>>>(tQ, tK, tV, tM,
                                                     route_len, src_len, RMAX, SMAX);
        }
        gemm(tM, x_Wo + oW, x_bo + oB, tO, MR, HID, HID, 0);
        add_norm_kernel<<<MR / 8, 256, 0, stream>>>(route_out, tO, rn2a + oB,
                                                    rn2b + oB, route_out2, MR);
        // ---- route FFN (no residual: route_emb = FFN(route_out2)) ----
        gemm(route_out2, rfW1 + oF, rfb1 + oB2, tM, MR, 2 * HID, HID, 1);
        gemm(tM, rfW2 + oF, rfb2 + oB, route_emb, MR, HID, 2 * HID, 0);
    }

    // outputs: route_outputs [RMAX,NB,HID] then hidden [1,NB,HID]
    write_route_outputs_kernel<<<(int)((SR + 255) / 256), 256, 0, stream>>>(
        route_emb, (float*)d_out);
    pool_fc_kernel<<<NB, HID, 0, stream>>>(route_emb, route_len, pro, temporal,
                                           fcW, fcb, (float*)d_out + SR);
}